// BERTEmbedding_65274912964883
// MI455X (gfx1250) — compile-verified
//
#include <hip/hip_runtime.h>

typedef __attribute__((ext_vector_type(2))) float v2f;
typedef __attribute__((ext_vector_type(8))) float v8f;
typedef __attribute__((ext_vector_type(4))) int   v4i;

#define EMBED        128
#define MAXLEN       200
#define NGR          21      // genre rows actually used (0..20)
#define KPAD         24      // K padded to multiple of 4 for WMMA f32 16x16x4
#define MAXG         8
#define TOK_PER_BLK  64      // 4 tiles of 16 tokens per block

__global__ __launch_bounds__(256)
void bert_embed_wmma_kernel(const int*   __restrict__ seq,      // [51200]
                            const float* __restrict__ tok_tab,  // [VOCAB*128]
                            const float* __restrict__ gen_tab,  // [21*128]
                            const float* __restrict__ pos_tab,  // [200*128]
                            const int*   __restrict__ tgids,    // [VOCAB*8]
                            const int*   __restrict__ gcnt,     // [VOCAB]
                            float*       __restrict__ out)      // [51200*128]
{
    __shared__ float sG[KPAD * EMBED];   // 12 KB padded genre table (rows 21..23 = 0)
    __shared__ int   sSeq[TOK_PER_BLK];  // vocab ids for this block's tokens

    const int tid        = threadIdx.x;
    const int blockStart = blockIdx.x * TOK_PER_BLK;

    // Stage genre table into LDS (pad rows with zeros so K=21..23 contribute nothing).
    for (int i = tid; i < KPAD * EMBED; i += 256)
        sG[i] = (i < NGR * EMBED) ? gen_tab[i] : 0.0f;
    if (tid < TOK_PER_BLK)
        sSeq[tid] = seq[blockStart + tid];
    __syncthreads();

    const int lane = tid & 31;
    const int wave = tid >> 5;
    const int tile = wave >> 1;              // 0..3: which 16-token tile
    const int half = wave & 1;               // 0..1: n-tiles 0-3 or 4-7
    const int tileBase = blockStart + tile * 16;

    // ---- Build A fragments (16x4 f32 per K-step, 6 steps for K=24) ----
    // lanes 0-15: token row = lane,  K sub-offset 0 (VGPR0=K0, VGPR1=K1)
    // lanes16-31: token row = lane-16, K sub-offset 2 (VGPR0=K2, VGPR1=K3)
    const int mrow = lane & 15;
    const int koff = (lane >> 4) * 2;

    const int vocab_id = sSeq[tile * 16 + mrow];
    v4i g0 = *(const v4i*)(tgids + (size_t)vocab_id * MAXG);
    v4i g1 = *(const v4i*)(tgids + (size_t)vocab_id * MAXG + 4);
    const int gid[MAXG] = {g0.x, g0.y, g0.z, g0.w, g1.x, g1.y, g1.z, g1.w};
    const int   cnt = gcnt[vocab_id];
    const float inv = 1.0f / (float)cnt;

    v2f A[6];
#pragma unroll
    for (int k = 0; k < 6; ++k) {
        const int tg0 = 4 * k + koff;
        const int tg1 = tg0 + 1;
        float w0 = 0.0f, w1 = 0.0f;
#pragma unroll
        for (int j = 0; j < MAXG; ++j) {
            const float valid = (j < cnt) ? 1.0f : 0.0f;   // duplicates sum, as in reference mask
            w0 += (gid[j] == tg0) ? valid : 0.0f;
            w1 += (gid[j] == tg1) ? valid : 0.0f;
        }
        A[k].x = w0 * inv;   // fold 1/count into the weights
        A[k].y = w1 * inv;
    }

    const int n      = lane & 15;           // N column for B/C/D
    const int rowAdd = (lane >> 4) * 8;     // C/D: upper lanes hold M = r+8

#pragma unroll
    for (int nt = 0; nt < 4; ++nt) {
        const int ntile = half * 4 + nt;
        const int dim   = ntile * 16 + n;

        v8f c = {};
#pragma unroll
        for (int k = 0; k < 6; ++k) {
            // B 4x16 f32: VGPR0 = row K=koff, VGPR1 = row K=koff+1, column = lane&15
            v2f b;
            b.x = sG[(4 * k + koff + 0) * EMBED + dim];
            b.y = sG[(4 * k + koff + 1) * EMBED + dim];
            c = __builtin_amdgcn_wmma_f32_16x16x4_f32(
                    /*neg_a=*/false, A[k], /*neg_b=*/false, b,
                    /*c_mod=*/(short)0, c, /*reuse_a=*/false, /*reuse_b=*/false);
        }

        // Add token + positional embeddings and store the 16x16 D tile.
#pragma unroll
        for (int r = 0; r < 8; ++r) {
            const int m   = r + rowAdd;
            const int t   = tileBase + m;            // flat token index
            const int vid = sSeq[tile * 16 + m];     // LDS broadcast read
            const int l   = t % MAXLEN;              // position row
            const float v = c[r]
                          + tok_tab[(size_t)vid * EMBED + dim]
                          + pos_tab[(size_t)l * EMBED + dim];
            out[(size_t)t * EMBED + dim] = v;
        }
    }
}

extern "C" void kernel_launch(void* const* d_in, const int* in_sizes, int n_in,
                              void* d_out, int out_size, void* d_ws, size_t ws_size,
                              hipStream_t stream) {
    const int*   seq     = (const int*)  d_in[0];   // (256,200)
    const float* tok_tab = (const float*)d_in[1];   // (100000,128)
    const float* gen_tab = (const float*)d_in[2];   // (21,128)
    const float* pos_tab = (const float*)d_in[3];   // (200,128)
    const int*   tgids   = (const int*)  d_in[4];   // (100000,8)
    const int*   gcnt    = (const int*)  d_in[5];   // (100000,)
    float*       out     = (float*)d_out;           // (256,200,128)

    const int total_tokens = in_sizes[0];           // 51200
    const int grid = total_tokens / TOK_PER_BLK;    // 800, exact

    bert_embed_wmma_kernel<<<grid, 256, 0, stream>>>(
        seq, tok_tab, gen_tab, pos_tab, tgids, gcnt, out);
}